// Model_20658792694235
// MI455X (gfx1250) — compile-verified
//
#include <hip/hip_runtime.h>
#include <hip/hip_bf16.h>
#include <math.h>

typedef float v2f __attribute__((ext_vector_type(2)));
typedef float v8f __attribute__((ext_vector_type(8)));
typedef int v4i_ __attribute__((vector_size(4 * sizeof(int))));

#define N_DIM 4096
#define S0_DIM 256
#define S1_DIM 128
#define S2_DIM 64
#define TAU_INV 2.0f
#define THRESH_V 0.9f
#define BPAD 260   // staged B-tile row stride (floats): 256 + 4 pad

static __device__ __forceinline__ float waveReduceSum(float v) {
  #pragma unroll
  for (int m = 16; m >= 1; m >>= 1) v += __shfl_xor(v, m, 32);
  return v;
}
// reduce within each 16-lane half of the wave
static __device__ __forceinline__ float halfReduceSum(float v) {
  #pragma unroll
  for (int m = 8; m >= 1; m >>= 1) v += __shfl_xor(v, m, 32);
  return v;
}

// ---- CDNA5 async global->LDS copy (ASYNCcnt-tracked) ----------------
static __device__ __forceinline__ void asyncLoadB128(const float* gsrc, float* ldst) {
#if __has_builtin(__builtin_amdgcn_global_load_async_to_lds_b128)
  __builtin_amdgcn_global_load_async_to_lds_b128(
      (__attribute__((address_space(1))) v4i_*)gsrc,
      (__attribute__((address_space(3))) v4i_*)ldst, 0, 0);
#else
  unsigned int loff =
      (unsigned int)(unsigned long long)(__attribute__((address_space(3))) void*)ldst;
  asm volatile("global_load_async_to_lds_b128 %0, %1, off"
               :: "v"(loff), "v"((unsigned long long)(size_t)gsrc)
               : "memory");
#endif
}
static __device__ __forceinline__ void waitAsync0() {
#if __has_builtin(__builtin_amdgcn_s_wait_asynccnt)
  __builtin_amdgcn_s_wait_asynccnt(0);
#else
  asm volatile("s_wait_asynccnt 0x0" ::: "memory");
#endif
}

// ---------------- small GEMM chain (negligible FLOPs) ----------------
__global__ void k_gemm(const float* __restrict__ A, const float* __restrict__ B,
                       float* __restrict__ C, int M, int N, int K) {
  int idx = blockIdx.x * blockDim.x + threadIdx.x;
  if (idx >= M * N) return;
  int i = idx / N, j = idx % N;
  float s = 0.f;
  for (int k = 0; k < K; ++k) s = fmaf(A[i * K + k], B[k * N + j], s);
  C[idx] = s;
}

// h1[i][j] = elu( sum_k V2[k][i] * fc1_w[j][k] + fc1_b[j] ),  i<N, j<S1
__global__ void k_h1(const float* __restrict__ V2, const float* __restrict__ w,
                     const float* __restrict__ b, float* __restrict__ h1) {
  int idx = blockIdx.x * blockDim.x + threadIdx.x;
  if (idx >= N_DIM * S1_DIM) return;
  int i = idx / S1_DIM, j = idx % S1_DIM;
  float s = b[j];
  for (int k = 0; k < S2_DIM; ++k)
    s = fmaf(V2[k * N_DIM + i], w[j * S2_DIM + k], s);
  h1[idx] = (s > 0.f) ? s : (expf(s) - 1.f);
}

// h[i][j] = sum_k h1[i][k]*fc2_w[j][k] + fc2_b[j]; hn = row-normalized h
__global__ void k_hn(const float* __restrict__ h1, const float* __restrict__ w,
                     const float* __restrict__ b, float* __restrict__ hn) {
  int i = blockIdx.x, j = threadIdx.x;
  float s = b[j];
  const float* hr = h1 + i * S1_DIM;
  const float* wr = w + j * S1_DIM;
  #pragma unroll 8
  for (int k = 0; k < S1_DIM; ++k) s = fmaf(hr[k], wr[k], s);
  __shared__ float red[256];
  red[j] = s * s;
  __syncthreads();
  for (int off = 128; off >= 1; off >>= 1) {
    if (j < off) red[j] += red[j + off];
    __syncthreads();
  }
  float norm = sqrtf(red[0]);
  hn[i * S0_DIM + j] = s / fmaxf(norm, 1e-12f);
}

// ---------------- loss1: sum((A - X@V2)^2), fused WMMA f32 K=64 ------
// wave strip: 16 rows x 256 cols. waves = 256 row-tiles * 16 col-groups.
__global__ void k_loss1(const float* __restrict__ X, const float* __restrict__ V2,
                        const float* __restrict__ Amat, float* __restrict__ acc) {
  const int lane = threadIdx.x & 31;
  const int w = blockIdx.x * 8 + (threadIdx.x >> 5);
  const int rt = w >> 4;          // 0..255
  const int cg = w & 15;          // 0..15
  const int row0 = rt * 16;
  const int colbase = cg * 256;
  const int m16 = lane & 15;
  const int khi = lane >> 4;

  // A-operand (X rows), K=64: a[ks] holds k = ks*4 + khi*2 .. +1
  v2f a[16];
  const float* xr = X + (row0 + m16) * S2_DIM;
  #pragma unroll
  for (int ks = 0; ks < 16; ++ks) {
    int k = ks * 4 + khi * 2;
    a[ks] = *(const v2f*)(xr + k);
  }

  float local = 0.f;
  for (int ct = 0; ct < 16; ++ct) {
    const int col0 = colbase + ct * 16;
    __builtin_prefetch(Amat + (size_t)(row0 + m16) * N_DIM + colbase + ((ct + 1) & 15) * 16, 0, 0);
    v8f c = {0.f, 0.f, 0.f, 0.f, 0.f, 0.f, 0.f, 0.f};
    #pragma unroll
    for (int ks = 0; ks < 16; ++ks) {
      int k = ks * 4 + khi * 2;
      v2f bv;
      bv.x = V2[k * N_DIM + col0 + m16];
      bv.y = V2[(k + 1) * N_DIM + col0 + m16];
      c = __builtin_amdgcn_wmma_f32_16x16x4_f32(false, a[ks], false, bv,
                                                (short)0, c, false, false);
    }
    #pragma unroll
    for (int r = 0; r < 8; ++r) {
      int m = r + 8 * khi;
      float d = Amat[(size_t)(row0 + m) * N_DIM + col0 + m16] - c[r];
      local = fmaf(d, d, local);
    }
  }
  local = waveReduceSum(local);
  if (lane == 0) atomicAdd(acc, local);
}

// ---------------- refl_sim strips: rowtotal + sim-masked pos ---------
// block: 128 rows (8 waves x 16) x 1024 cols, K=256.
// B tiles (16 cols x 256 K = contiguous 16KB of hn) are async-staged
// into LDS once per block (8x L2 traffic cut) with double buffering.
__global__ void k_refl(const float* __restrict__ hn, const float* __restrict__ sim,
                       float* __restrict__ rowtot, float* __restrict__ possim) {
  __shared__ float sB[2][16 * BPAD];
  const int tid = threadIdx.x;
  const int lane = tid & 31;
  const int wv = tid >> 5;
  const int rb = blockIdx.x >> 2;   // 0..31
  const int cg = blockIdx.x & 3;    // 0..3
  const int row0 = rb * 128 + wv * 16;
  const int colbase = cg * 1024;
  const int m16 = lane & 15;
  const int khi = lane >> 4;

  // A-operand: hn rows, K=256 -> 64 x v2f held in registers
  v2f a[64];
  const float* ar = hn + (row0 + m16) * S0_DIM;
  #pragma unroll
  for (int ks = 0; ks < 64; ++ks) {
    int k = ks * 4 + khi * 2;
    a[ks] = *(const v2f*)(ar + k);
  }

  float atot[8] = {0.f, 0.f, 0.f, 0.f, 0.f, 0.f, 0.f, 0.f};
  float apos[8] = {0.f, 0.f, 0.f, 0.f, 0.f, 0.f, 0.f, 0.f};

  // preload tile 0 (each thread issues 4 x b128 = 16KB/block)
  {
    const float* g = hn + (size_t)colbase * S0_DIM;
    #pragma unroll
    for (int q = 0; q < 4; ++q) {
      int f = (tid + 256 * q) * 4;
      asyncLoadB128(g + f, &sB[0][(f >> 8) * BPAD + (f & 255)]);
    }
  }
  waitAsync0();
  __syncthreads();

  for (int ct = 0; ct < 64; ++ct) {
    const int cur = ct & 1;
    const int col0 = colbase + ct * 16;
    if (ct + 1 < 64) {   // issue next tile into the other buffer
      const float* g = hn + (size_t)(colbase + (ct + 1) * 16) * S0_DIM;
      #pragma unroll
      for (int q = 0; q < 4; ++q) {
        int f = (tid + 256 * q) * 4;
        asyncLoadB128(g + f, &sB[cur ^ 1][(f >> 8) * BPAD + (f & 255)]);
      }
    }
    const float* br = &sB[cur][m16 * BPAD];   // B[k][n] = hn[col0+n][k]
    v8f c = {0.f, 0.f, 0.f, 0.f, 0.f, 0.f, 0.f, 0.f};
    #pragma unroll
    for (int ks = 0; ks < 64; ++ks) {
      int k = ks * 4 + khi * 2;
      v2f bv = *(const v2f*)(br + k);
      c = __builtin_amdgcn_wmma_f32_16x16x4_f32(false, a[ks], false, bv,
                                                (short)0, c, false, false);
    }
    #pragma unroll
    for (int r = 0; r < 8; ++r) {
      int m = r + 8 * khi;
      float e = expf(c[r] * TAU_INV);
      atot[r] += e;
      if (sim[(size_t)(row0 + m) * N_DIM + col0 + m16] > THRESH_V) apos[r] += e;
    }
    waitAsync0();      // next tile landed in LDS
    __syncthreads();   // visible to all 8 waves; prev buffer free
  }

  #pragma unroll
  for (int r = 0; r < 8; ++r) {
    atot[r] = halfReduceSum(atot[r]);
    apos[r] = halfReduceSum(apos[r]);
  }
  if (m16 == 0) {
    #pragma unroll
    for (int r = 0; r < 8; ++r) {
      int row = row0 + r + 8 * khi;
      atomicAdd(&rowtot[row], atot[r]);
      atomicAdd(&possim[row], apos[r]);
    }
  }
}

// ---------------- clique/neighbor positive sums (gathered, deduped) --
__global__ void k_pos_gather(const float* __restrict__ hn, const int* __restrict__ idx,
                             int K, float* __restrict__ pos) {
  const int lane = threadIdx.x & 31;
  const int row = blockIdx.x * 8 + (threadIdx.x >> 5);
  float myA[8];
  const float* ar = hn + (size_t)row * S0_DIM + lane * 8;
  #pragma unroll
  for (int t = 0; t < 8; ++t) myA[t] = ar[t];

  int myIdx = (lane < K) ? idx[row * K + lane] : -1;
  float psum = 0.f;
  for (int k = 0; k < K; ++k) {
    int j = __shfl(myIdx, k, 32);
    bool dup = false;                    // mask has set semantics: dedupe
    for (int l = 0; l < k; ++l) dup = dup || (__shfl(myIdx, l, 32) == j);
    if (dup) continue;
    const float* brr = hn + (size_t)j * S0_DIM + lane * 8;
    float d = 0.f;
    #pragma unroll
    for (int t = 0; t < 8; ++t) d = fmaf(myA[t], brr[t], d);
    d = waveReduceSum(d);
    psum += expf(d * TAU_INV);
  }
  if (lane == 0) pos[row] = psum;
}

// ---------------- loss5: sum(min(x,0)^2) reductions -------------------
__global__ void k_neg_sq(const float* __restrict__ X, int n, float* __restrict__ acc) {
  int i = blockIdx.x * blockDim.x + threadIdx.x;
  float v = 0.f;
  for (; i < n; i += gridDim.x * blockDim.x) {
    float m = fminf(X[i], 0.f);
    v = fmaf(m, m, v);
  }
  v = waveReduceSum(v);
  __shared__ float red[8];
  int lane = threadIdx.x & 31, wv = threadIdx.x >> 5;
  if (lane == 0) red[wv] = v;
  __syncthreads();
  if (threadIdx.x == 0) {
    float s = 0.f;
    for (int t = 0; t < 8; ++t) s += red[t];
    atomicAdd(acc, s);
  }
}

// ---------------- finalize: 6 scalar outputs -------------------------
__global__ void k_final(const float* __restrict__ rowtot, const float* __restrict__ pcl,
                        const float* __restrict__ pne, const float* __restrict__ psim,
                        const float* __restrict__ scal, float* __restrict__ out) {
  int tid = threadIdx.x;
  float s2 = 0.f, s3 = 0.f, s4 = 0.f;
  for (int i = tid; i < N_DIM; i += 256) {
    float rt = rowtot[i];
    float c2 = pcl[i] / rt;   s2 += (c2 == 0.f) ? 0.f : -logf(c2);
    float c3 = pne[i] / rt;   s3 += (c3 == 0.f) ? 0.f : -logf(c3);
    float c4 = psim[i] / rt;  s4 += (c4 == 0.f) ? 0.f : -logf(c4);
  }
  __shared__ float r2[256], r3[256], r4[256];
  r2[tid] = s2; r3[tid] = s3; r4[tid] = s4;
  __syncthreads();
  for (int off = 128; off >= 1; off >>= 1) {
    if (tid < off) { r2[tid] += r2[tid + off]; r3[tid] += r3[tid + off]; r4[tid] += r4[tid + off]; }
    __syncthreads();
  }
  if (tid == 0) {
    float l1 = scal[0], l5 = scal[1];
    float l2 = r2[0] / (float)N_DIM;
    float l3 = r3[0] / (float)N_DIM;
    float l4 = r4[0] / (float)N_DIM;
    float total = l1 + 0.1f * (l2 + l3 + l4) + l5;
    out[0] = total; out[1] = l1; out[2] = l2; out[3] = l3; out[4] = l4; out[5] = l5;
  }
}

extern "C" void kernel_launch(void* const* d_in, const int* in_sizes, int n_in,
                              void* d_out, int out_size, void* d_ws, size_t ws_size,
                              hipStream_t stream) {
  const float* A    = (const float*)d_in[0];
  const float* U0   = (const float*)d_in[1];
  const float* U1   = (const float*)d_in[2];
  const float* U2   = (const float*)d_in[3];
  const float* V2   = (const float*)d_in[4];
  const float* fc1w = (const float*)d_in[5];
  const float* fc1b = (const float*)d_in[6];
  const float* fc2w = (const float*)d_in[7];
  const float* fc2b = (const float*)d_in[8];
  const float* simm = (const float*)d_in[9];
  const int*   cli  = (const int*)d_in[10];
  const int*   nei  = (const int*)d_in[11];

  float* ws = (float*)d_ws;
  float* W1     = ws;                         // 4096*128
  float* X      = W1 + 4096 * 128;            // 4096*64
  float* h1     = X + 4096 * 64;              // 4096*128
  float* hn     = h1 + 4096 * 128;            // 4096*256
  float* rowtot = hn + 4096 * 256;            // 4096
  float* psim   = rowtot + 4096;              // 4096
  float* pcl    = psim + 4096;                // 4096
  float* pne    = pcl + 4096;                 // 4096
  float* scal   = pne + 4096;                 // [0]=loss1 acc, [1]=loss5 acc

  // zero the accumulated buffers (rowtot, psim, pcl, pne, scal)
  (void)hipMemsetAsync(rowtot, 0, (4 * 4096 + 2) * sizeof(float), stream);

  // factor chain
  k_gemm<<<(4096 * 128 + 255) / 256, 256, 0, stream>>>(U0, U1, W1, 4096, 128, 256);
  k_gemm<<<(4096 * 64 + 255) / 256, 256, 0, stream>>>(W1, U2, X, 4096, 64, 128);
  // projection
  k_h1<<<(4096 * 128 + 255) / 256, 256, 0, stream>>>(V2, fc1w, fc1b, h1);
  k_hn<<<4096, 256, 0, stream>>>(h1, fc2w, fc2b, hn);
  // fused big GEMMs
  k_loss1<<<512, 256, 0, stream>>>(X, V2, A, &scal[0]);
  k_refl<<<128, 256, 0, stream>>>(hn, simm, rowtot, psim);
  // gathered positives
  k_pos_gather<<<512, 256, 0, stream>>>(hn, cli, 8, pcl);
  k_pos_gather<<<512, 256, 0, stream>>>(hn, nei, 16, pne);
  // nonnegativity penalty
  k_neg_sq<<<256, 256, 0, stream>>>(U0, 4096 * 256, &scal[1]);
  k_neg_sq<<<64, 256, 0, stream>>>(U1, 256 * 128, &scal[1]);
  k_neg_sq<<<16, 256, 0, stream>>>(U2, 128 * 64, &scal[1]);
  k_neg_sq<<<64, 256, 0, stream>>>(V2, 64 * 4096, &scal[1]);
  // finalize
  k_final<<<1, 256, 0, stream>>>(rowtot, pcl, pne, psim, scal, (float*)d_out);
}